// GCN_3478923510362
// MI455X (gfx1250) — compile-verified
//
#include <hip/hip_runtime.h>

// ---------------------------------------------------------------------------
// GCN forward for MI455X (gfx1250, wave32).
//  - GEMMs use V_WMMA_F32_16X16X4_F32 (full fp32 precision).
//  - Graph aggregation uses a CSR built once per launch (no fp atomics in the
//    per-layer hot path), gathers are L2-resident (51 MB << 192 MB L2).
// ---------------------------------------------------------------------------

typedef __attribute__((ext_vector_type(2))) float v2f;
typedef __attribute__((ext_vector_type(8))) float v8f;

#define H 128
#define LDS_STRIDE 132   // 128 + 4 pad: avoids 16-way LDS bank conflicts

// ---------------------------------------------------------------------------
// WMMA GEMM: C[N,128] = A[N,128] * W[128,128] (+bias, +relu when BIAS_RELU)
// grid = N/16 blocks, 256 threads (8 waves). Wave w computes cols [16w,16w+16).
// ---------------------------------------------------------------------------
template <bool BIAS_RELU>
__launch_bounds__(256)
__global__ void gemm128_wmma(const float* __restrict__ A,
                             const float* __restrict__ W,
                             const float* __restrict__ bias,
                             float* __restrict__ C)
{
    __shared__ float As[16 * LDS_STRIDE];

    const int row0 = blockIdx.x * 16;
    const int t    = threadIdx.x;

    // Cooperative load of the 16x128 A tile (each thread copies 8 floats).
    {
        const int r = t >> 4;            // 0..15
        const int c = (t & 15) * 8;      // 0,8,...,120
        const float4* srcp = (const float4*)(A + (size_t)(row0 + r) * H + c);
        float4* dstp = (float4*)(&As[r * LDS_STRIDE + c]);
        dstp[0] = srcp[0];
        dstp[1] = srcp[1];
    }
    __syncthreads();

    const int wave  = t >> 5;            // 0..7 -> output column tile
    const int lane  = t & 31;
    const int n0    = wave * 16;
    const int mrow  = lane & 15;
    const int khalf = (lane >> 4) * 2;   // 0 (lanes 0-15) or 2 (lanes 16-31)
    const int ncol  = n0 + (lane & 15);

    // A fragment base: lane reads A[mrow][k0+khalf], A[mrow][k0+khalf+1]
    const float* arow  = &As[mrow * LDS_STRIDE + khalf];
    // B fragment base: lane reads W[k0+khalf][ncol], W[k0+khalf+1][ncol]
    const float* bbase = W + khalf * H + ncol;

    v8f acc = {};
#pragma unroll
    for (int k0 = 0; k0 < H; k0 += 4) {
        v2f a, b;
        a.x = arow[k0];
        a.y = arow[k0 + 1];
        b.x = bbase[k0 * H];
        b.y = bbase[k0 * H + H];
        acc = __builtin_amdgcn_wmma_f32_16x16x4_f32(
            false, a, false, b, (short)0, acc, false, false);
    }

    // C layout: VGPR r -> row (r + 8*(lane>=16)), col = ncol
    const int rbase = row0 + (lane >> 4) * 8;
    if (BIAS_RELU) {
        const float bv = bias[ncol];
#pragma unroll
        for (int r = 0; r < 8; ++r)
            C[(size_t)(rbase + r) * H + ncol] = fmaxf(acc[r] + bv, 0.0f);
    } else {
#pragma unroll
        for (int r = 0; r < 8; ++r)
            C[(size_t)(rbase + r) * H + ncol] = acc[r];
    }
}

// ---------------------------------------------------------------------------
// Final projection: out[N,16] = h[N,128] * Wo[128,16] + bo.
// One wave per 16-row tile; 8 tiles per 256-thread block.
// ---------------------------------------------------------------------------
__launch_bounds__(256)
__global__ void gemm_out_wmma(const float* __restrict__ A,
                              const float* __restrict__ Wo,
                              const float* __restrict__ bo,
                              float* __restrict__ C,
                              int nTiles)
{
    const int wave = (blockIdx.x * blockDim.x + threadIdx.x) >> 5;
    if (wave >= nTiles) return;          // wave-uniform: EXEC stays all-1s

    const int lane  = threadIdx.x & 31;
    const int row0  = wave * 16;
    const int mrow  = lane & 15;
    const int khalf = (lane >> 4) * 2;
    const int ncol  = lane & 15;

    const float* arow  = A + (size_t)(row0 + mrow) * H + khalf;
    const float* bbase = Wo + khalf * 16 + ncol;

    v8f acc = {};
#pragma unroll
    for (int k0 = 0; k0 < H; k0 += 4) {
        v2f a, b;
        a.x = arow[k0];
        a.y = arow[k0 + 1];
        b.x = bbase[k0 * 16];
        b.y = bbase[k0 * 16 + 16];
        acc = __builtin_amdgcn_wmma_f32_16x16x4_f32(
            false, a, false, b, (short)0, acc, false, false);
    }

    const float bv = bo[ncol];
    const int rbase = row0 + (lane >> 4) * 8;
#pragma unroll
    for (int r = 0; r < 8; ++r)
        C[(size_t)(rbase + r) * 16 + ncol] = acc[r] + bv;
}

// ---------------------------------------------------------------------------
// Graph preprocessing kernels (run once per launch)
// ---------------------------------------------------------------------------
__global__ void k_init_deg(float* deg, int* cnt, int n)
{
    int i = blockIdx.x * blockDim.x + threadIdx.x;
    if (i < n) { deg[i] = 1.0f; cnt[i] = 0; }   // self-loop fill weight 1.0
}

__global__ void k_edge_count(const int* __restrict__ dst,
                             const float* __restrict__ ew,
                             float* deg, int* cnt, int nE)
{
    int e = blockIdx.x * blockDim.x + threadIdx.x;
    if (e < nE) {
        int d = dst[e];
        atomicAdd(&deg[d], ew[e]);
        atomicAdd(&cnt[d], 1);
    }
}

__global__ void k_rsqrt(float* v, int n)
{
    int i = blockIdx.x * blockDim.x + threadIdx.x;
    if (i < n) v[i] = rsqrtf(v[i]);
}

#define SCAN_B 256

__global__ void k_block_reduce(const int* __restrict__ cnt, int* bsum, int n)
{
    __shared__ int sm[SCAN_B];
    int i = blockIdx.x * SCAN_B + threadIdx.x;
    sm[threadIdx.x] = (i < n) ? cnt[i] : 0;
    __syncthreads();
    for (int s = SCAN_B / 2; s > 0; s >>= 1) {
        if (threadIdx.x < s) sm[threadIdx.x] += sm[threadIdx.x + s];
        __syncthreads();
    }
    if (threadIdx.x == 0) bsum[blockIdx.x] = sm[0];
}

__global__ void k_scan_bsums(int* bsum, int nb, int* rowptr, int N, int E)
{
    __shared__ int sm[1024];
    int t = threadIdx.x;
    sm[t] = (t < nb) ? bsum[t] : 0;
    __syncthreads();
    for (int off = 1; off < 1024; off <<= 1) {
        int v = (t >= off) ? sm[t - off] : 0;
        __syncthreads();
        sm[t] += v;
        __syncthreads();
    }
    if (t < nb) bsum[t] = (t == 0) ? 0 : sm[t - 1];  // exclusive
    if (t == 0) rowptr[N] = E;
}

__global__ void k_scan_write(const int* __restrict__ cnt,
                             const int* __restrict__ bsum,
                             int* rowptr, int* cursor, int n)
{
    __shared__ int sm[SCAN_B];
    int t = threadIdx.x;
    int i = blockIdx.x * SCAN_B + t;
    int v = (i < n) ? cnt[i] : 0;
    sm[t] = v;
    __syncthreads();
    for (int off = 1; off < SCAN_B; off <<= 1) {
        int u = (t >= off) ? sm[t - off] : 0;
        __syncthreads();
        sm[t] += u;
        __syncthreads();
    }
    int excl = sm[t] - v + bsum[blockIdx.x];
    if (i < n) { rowptr[i] = excl; cursor[i] = excl; }
}

__global__ void k_edge_place(const int* __restrict__ src,
                             const int* __restrict__ dst,
                             const float* __restrict__ ew,
                             const float* __restrict__ dinv,
                             int* cursor,
                             int* __restrict__ e_src,
                             float* __restrict__ e_w,
                             int nE)
{
    int e = blockIdx.x * blockDim.x + threadIdx.x;
    if (e < nE) {
        int s = src[e], d = dst[e];
        int pos = atomicAdd(&cursor[d], 1);
        e_src[pos] = s;
        e_w[pos]   = dinv[s] * ew[e] * dinv[d];   // GCN edge norm
    }
}

// ---------------------------------------------------------------------------
// Fused per-layer aggregation: one wave per node, lane owns 4 features.
//   h_out[d] = relu( bias + self_norm*hw[d] + sum_{e: dst==d} norm_e*hw[src_e] )
// Pure gather (no atomics); hw (51 MB) is L2-resident on MI455X.
// ---------------------------------------------------------------------------
__launch_bounds__(256)
__global__ void k_aggregate(const float* __restrict__ hw,
                            const int* __restrict__ rowptr,
                            const int* __restrict__ e_src,
                            const float* __restrict__ e_w,
                            const float* __restrict__ dinv,
                            const float* __restrict__ bias,
                            float* __restrict__ hout,
                            int n)
{
    int g = blockIdx.x * blockDim.x + threadIdx.x;
    int node = g >> 5;
    if (node >= n) return;
    int lane = g & 31;
    int f = lane * 4;

    float dv = dinv[node];
    float selfw = dv * dv;

    const float4 hv = *(const float4*)(hw + (size_t)node * H + f);
    const float4 bv = *(const float4*)(bias + f);
    float4 acc;
    acc.x = fmaf(hv.x, selfw, bv.x);
    acc.y = fmaf(hv.y, selfw, bv.y);
    acc.z = fmaf(hv.z, selfw, bv.z);
    acc.w = fmaf(hv.w, selfw, bv.w);

    int beg = rowptr[node], end = rowptr[node + 1];
    for (int j = beg; j < end; j += 32) {
        int idx = j + lane;
        int   sL = (idx < end) ? e_src[idx] : 0;
        float wL = (idx < end) ? e_w[idx] : 0.0f;
        int m = end - j; if (m > 32) m = 32;
        for (int u = 0; u < m; ++u) {
            int   s = __shfl(sL, u);
            float w = __shfl(wL, u);
            const float4 v = *(const float4*)(hw + (size_t)s * H + f);
            acc.x = fmaf(v.x, w, acc.x);
            acc.y = fmaf(v.y, w, acc.y);
            acc.z = fmaf(v.z, w, acc.z);
            acc.w = fmaf(v.w, w, acc.w);
        }
    }

    float4 r;
    r.x = fmaxf(acc.x, 0.0f);
    r.y = fmaxf(acc.y, 0.0f);
    r.z = fmaxf(acc.z, 0.0f);
    r.w = fmaxf(acc.w, 0.0f);
    *(float4*)(hout + (size_t)node * H + f) = r;
}

// ---------------------------------------------------------------------------
// Launch
// ---------------------------------------------------------------------------
extern "C" void kernel_launch(void* const* d_in, const int* in_sizes, int n_in,
                              void* d_out, int out_size, void* d_ws, size_t ws_size,
                              hipStream_t stream)
{
    const float* x    = (const float*)d_in[0];
    const int*   eidx = (const int*)  d_in[1];
    const float* ew   = (const float*)d_in[2];
    const float* Wf   = (const float*)d_in[3];
    const float* bf   = (const float*)d_in[4];
    const float* W1   = (const float*)d_in[5];
    const float* b1   = (const float*)d_in[6];
    const float* W2   = (const float*)d_in[7];
    const float* b2   = (const float*)d_in[8];
    const float* Wo   = (const float*)d_in[9];
    const float* bo   = (const float*)d_in[10];
    float* out = (float*)d_out;

    const int N = in_sizes[0] / H;   // 100000 (multiple of 16)
    const int E = in_sizes[1] / 2;   // 1600000
    const int* esrc = eidx;
    const int* edst = eidx + E;

    // Carve workspace (256B-aligned)
    char* p = (char*)d_ws;
    auto carve = [&](size_t bytes) -> char* {
        char* r = p;
        p += (bytes + 255) & ~(size_t)255;
        return r;
    };
    float* h      = (float*)carve((size_t)N * H * 4);
    float* hwbuf  = (float*)carve((size_t)N * H * 4);
    float* dinv   = (float*)carve((size_t)N * 4);       // deg -> rsqrt in place
    int*   cnt    = (int*)  carve((size_t)N * 4);
    int*   rowptr = (int*)  carve((size_t)(N + 1) * 4);
    int*   cursor = (int*)  carve((size_t)N * 4);
    int*   bsum   = (int*)  carve(4096);
    int*   es     = (int*)  carve((size_t)E * 4);
    float* ewn    = (float*)carve((size_t)E * 4);

    const int nBlkN = (N + 255) / 256;
    const int nBlkE = (E + 255) / 256;
    const int nb    = (N + SCAN_B - 1) / SCAN_B;        // 391 (< 1024)

    // --- graph preprocessing: degrees, norms, CSR by destination ---
    k_init_deg   <<<nBlkN, 256, 0, stream>>>(dinv, cnt, N);
    k_edge_count <<<nBlkE, 256, 0, stream>>>(edst, ew, dinv, cnt, E);
    k_rsqrt      <<<nBlkN, 256, 0, stream>>>(dinv, N);
    k_block_reduce<<<nb, SCAN_B, 0, stream>>>(cnt, bsum, N);
    k_scan_bsums <<<1, 1024, 0, stream>>>(bsum, nb, rowptr, N, E);
    k_scan_write <<<nb, SCAN_B, 0, stream>>>(cnt, bsum, rowptr, cursor, N);
    k_edge_place <<<nBlkE, 256, 0, stream>>>(esrc, edst, ew, dinv, cursor, es, ewn, E);

    const int rowTiles = N / 16;                        // 6250
    const int aggBlks  = (N * 32 + 255) / 256;

    // --- h = relu(x @ W_first + b_first) ---
    gemm128_wmma<true><<<rowTiles, 256, 0, stream>>>(x, Wf, bf, h);

    // --- GCN layer 1 ---
    gemm128_wmma<false><<<rowTiles, 256, 0, stream>>>(h, W1, nullptr, hwbuf);
    k_aggregate<<<aggBlks, 256, 0, stream>>>(hwbuf, rowptr, es, ewn, dinv, b1, h, N);

    // --- GCN layer 2 ---
    gemm128_wmma<false><<<rowTiles, 256, 0, stream>>>(h, W2, nullptr, hwbuf);
    k_aggregate<<<aggBlks, 256, 0, stream>>>(hwbuf, rowptr, es, ewn, dinv, b2, h, N);

    // --- out = h @ W_out + b_out ---
    gemm_out_wmma<<<(rowTiles + 7) / 8, 256, 0, stream>>>(h, Wo, bo, out, rowTiles);
}